// GATEncoder_7713761264112
// MI455X (gfx1250) — compile-verified
//
#include <hip/hip_runtime.h>
#include <math.h>

typedef __attribute__((ext_vector_type(2))) float v2f;
typedef __attribute__((ext_vector_type(8))) float v8f;

#define NEG_SLOPE 0.2f
#define GAT_EPS 1e-16f

// ---------------- utility fills ----------------
__global__ void fill_f32(float* __restrict__ p, float v, int n) {
    int i = blockIdx.x * blockDim.x + threadIdx.x;
    if (i < n) p[i] = v;
}

// ---------------- self-loop attr (fill_value='mean') ----------------
__global__ void loop_accum(const int* __restrict__ dst, const float* __restrict__ ea,
                           float* __restrict__ cnt, float* __restrict__ sum, int E) {
    int i = blockIdx.x * blockDim.x + threadIdx.x;
    if (i < E) {
        int d = dst[i];
        atomicAdd(&cnt[d], 1.0f);
        atomicAdd(&sum[d], ea[i]);
    }
}
__global__ void loop_final(float* __restrict__ sum, const float* __restrict__ cnt, int N) {
    int i = blockIdx.x * blockDim.x + threadIdx.x;
    if (i < N) sum[i] = sum[i] / fmaxf(cnt[i], 1.0f);
}

// ---------------- layer-1 GEMM: H1[N,128] = X[N,64] @ W[64,128] via f32 WMMA ----------------
// one wave (32 lanes) per 16x16 output tile; K=64 in steps of 4
__global__ void gemm_wmma_f32(const float* __restrict__ X, const float* __restrict__ W,
                              float* __restrict__ H, int N) {
    int lane = threadIdx.x;            // 0..31
    int half = lane >> 4;              // 0 or 1
    int l    = lane & 15;
    int m0 = blockIdx.y * 16;
    int n0 = blockIdx.x * 16;
    int arow = m0 + l; if (arow >= N) arow = N - 1;   // clamp reads for tail tiles
    int bcol = n0 + l;
    v8f c = {0.f, 0.f, 0.f, 0.f, 0.f, 0.f, 0.f, 0.f};
    for (int k = 0; k < 64; k += 4) {
        int ka = k + half * 2;         // A: lanes0-15 K={k,k+1}, lanes16-31 K={k+2,k+3}
        v2f a, b;
        a.x = X[arow * 64 + ka];
        a.y = X[arow * 64 + ka + 1];
        b.x = W[ka * 128 + bcol];      // B: lanes0-15 rows {k,k+1}, lanes16-31 rows {k+2,k+3}
        b.y = W[(ka + 1) * 128 + bcol];
        c = __builtin_amdgcn_wmma_f32_16x16x4_f32(false, a, false, b, (short)0, c, false, false);
    }
    int mrow = m0 + half * 8;          // C/D: VGPR r -> M = mrow + r, N = n0 + l
    for (int r = 0; r < 8; ++r) {
        int m = mrow + r;
        if (m < N) H[(size_t)m * 128 + n0 + l] = c[r];
    }
}

// ---------------- per-node attention logits, layer 1 ----------------
__global__ void node_att1(const float* __restrict__ H1, const float* __restrict__ att_src,
                          const float* __restrict__ att_dst, float* __restrict__ a_src,
                          float* __restrict__ a_dst, int N) {
    int i = blockIdx.x * blockDim.x + threadIdx.x;   // over N*8
    if (i >= N * 8) return;
    int n = i >> 3, h = i & 7;
    const float* hp = H1 + (size_t)n * 128 + h * 16;
    float s = 0.f, d = 0.f;
    for (int cc = 0; cc < 16; ++cc) {
        s = fmaf(hp[cc], att_src[h * 16 + cc], s);
        d = fmaf(hp[cc], att_dst[h * 16 + cc], d);
    }
    a_src[i] = s;
    a_dst[i] = d;
}

// ---------------- edge-attr constants: ce[h] = dot(We1_h, att_edge1_h), ce[8] = dot(We2, att_edge2) ----
__global__ void edge_consts(const float* __restrict__ We1, const float* __restrict__ ae1,
                            const float* __restrict__ We2, const float* __restrict__ ae2,
                            float* __restrict__ ce) {
    int h = threadIdx.x;
    if (h < 8) {
        float s = 0.f;
        for (int cc = 0; cc < 16; ++cc) s = fmaf(We1[h * 16 + cc], ae1[h * 16 + cc], s);
        ce[h] = s;
    } else if (h == 8) {
        float s = 0.f;
        for (int cc = 0; cc < 8; ++cc) s = fmaf(We2[cc], ae2[cc], s);
        ce[8] = s;
    }
}

// ---------------- layer-1 edge: leaky_relu logits + segment max ----------------
__global__ void edge_alpha1(const int* __restrict__ src, const int* __restrict__ dst,
                            const float* __restrict__ ea, const float* __restrict__ loopa,
                            const float* __restrict__ a_src, const float* __restrict__ a_dst,
                            const float* __restrict__ ce, float* __restrict__ alpha,
                            float* __restrict__ amax, int E, int Et) {
    int i = blockIdx.x * blockDim.x + threadIdx.x;
    if (i >= Et) return;
    int s, d; float eav;
    if (i < E) { s = src[i]; d = dst[i]; eav = ea[i]; }
    else       { s = d = i - E; eav = loopa[s]; }
    for (int h = 0; h < 8; ++h) {
        float a = a_src[s * 8 + h] + a_dst[d * 8 + h] + eav * ce[h];
        a = a > 0.f ? a : NEG_SLOPE * a;
        alpha[(size_t)i * 8 + h] = a;
        atomicMax(&amax[d * 8 + h], a);   // global_atomic_max_num_f32
    }
}

// ---------------- layer-1 edge: exp + segment sum ----------------
__global__ void edge_exp1(const int* __restrict__ dst, float* __restrict__ alpha,
                          const float* __restrict__ amax, float* __restrict__ den,
                          int E, int Et) {
    int i = blockIdx.x * blockDim.x + threadIdx.x;
    if (i >= Et) return;
    int d = (i < E) ? dst[i] : (i - E);
    for (int h = 0; h < 8; ++h) {
        float w = __expf(alpha[(size_t)i * 8 + h] - amax[d * 8 + h]);
        alpha[(size_t)i * 8 + h] = w;
        atomicAdd(&den[d * 8 + h], w);
    }
}

// ---------------- layer-1 edge: normalized scatter-add (one 128-thr block per edge) ----------------
__global__ void edge_agg1(const int* __restrict__ src, const int* __restrict__ dst,
                          const float* __restrict__ alpha, const float* __restrict__ den,
                          const float* __restrict__ H1, float* __restrict__ acc, int E) {
    int i = blockIdx.x;
    int t = threadIdx.x;               // 0..127 = h*16+c
    int s, d;
    if (i < E) { s = src[i]; d = dst[i]; } else { s = d = i - E; }
    int h = t >> 4;
    float w = alpha[(size_t)i * 8 + h] / (den[d * 8 + h] + GAT_EPS);
    atomicAdd(&acc[(size_t)d * 128 + t], w * H1[(size_t)s * 128 + t]);
}

// ---------------- layer-1 tail: head-mean + bias + ReLU ----------------
__global__ void node_out1(const float* __restrict__ acc, const float* __restrict__ b1,
                          float* __restrict__ H2, int N) {
    int i = blockIdx.x * blockDim.x + threadIdx.x;   // over N*16
    if (i >= N * 16) return;
    int n = i >> 4, cc = i & 15;
    float s = 0.f;
    for (int h = 0; h < 8; ++h) s += acc[(size_t)n * 128 + h * 16 + cc];
    s = s * 0.125f + b1[cc];
    H2[i] = fmaxf(s, 0.f);
}

// ---------------- layer-2 node: G2 = H2 @ W2, plus src/dst logits ----------------
__global__ void node_l2(const float* __restrict__ H2, const float* __restrict__ W2,
                        const float* __restrict__ as2, const float* __restrict__ ad2,
                        float* __restrict__ G2, float* __restrict__ a_src2,
                        float* __restrict__ a_dst2, int N) {
    int n = blockIdx.x * blockDim.x + threadIdx.x;
    if (n >= N) return;
    float g[8];
    for (int o = 0; o < 8; ++o) {
        float s = 0.f;
        for (int k = 0; k < 16; ++k) s = fmaf(H2[n * 16 + k], W2[k * 8 + o], s);
        g[o] = s;
    }
    float s = 0.f, d = 0.f;
    for (int o = 0; o < 8; ++o) {
        G2[(size_t)n * 8 + o] = g[o];
        s = fmaf(g[o], as2[o], s);
        d = fmaf(g[o], ad2[o], d);
    }
    a_src2[n] = s;
    a_dst2[n] = d;
}

// ---------------- layer-2 edge kernels (H=1) ----------------
__global__ void edge_alpha2(const int* __restrict__ src, const int* __restrict__ dst,
                            const float* __restrict__ ea, const float* __restrict__ loopa,
                            const float* __restrict__ a_src, const float* __restrict__ a_dst,
                            const float* __restrict__ ce, float* __restrict__ alpha,
                            float* __restrict__ amax, int E, int Et) {
    int i = blockIdx.x * blockDim.x + threadIdx.x;
    if (i >= Et) return;
    int s, d; float eav;
    if (i < E) { s = src[i]; d = dst[i]; eav = ea[i]; }
    else       { s = d = i - E; eav = loopa[s]; }
    float a = a_src[s] + a_dst[d] + eav * ce[8];
    a = a > 0.f ? a : NEG_SLOPE * a;
    alpha[i] = a;
    atomicMax(&amax[d], a);
}

__global__ void edge_exp2(const int* __restrict__ dst, float* __restrict__ alpha,
                          const float* __restrict__ amax, float* __restrict__ den,
                          int E, int Et) {
    int i = blockIdx.x * blockDim.x + threadIdx.x;
    if (i >= Et) return;
    int d = (i < E) ? dst[i] : (i - E);
    float w = __expf(alpha[i] - amax[d]);
    alpha[i] = w;
    atomicAdd(&den[d], w);
}

__global__ void edge_agg2(const int* __restrict__ src, const int* __restrict__ dst,
                          const float* __restrict__ alpha, const float* __restrict__ den,
                          const float* __restrict__ G2, float* __restrict__ acc,
                          int E, int Et) {
    int t = blockIdx.x * blockDim.x + threadIdx.x;   // over Et*8
    if (t >= Et * 8) return;
    int i = t >> 3, cc = t & 7;
    int s, d;
    if (i < E) { s = src[i]; d = dst[i]; } else { s = d = i - E; }
    float w = alpha[i] / (den[d] + GAT_EPS);
    atomicAdd(&acc[(size_t)d * 8 + cc], w * G2[(size_t)s * 8 + cc]);
}

__global__ void final_out(const float* __restrict__ acc, const float* __restrict__ b2,
                          float* __restrict__ out, int N) {
    int i = blockIdx.x * blockDim.x + threadIdx.x;   // over N*8
    if (i >= N * 8) return;
    out[i] = acc[i] + b2[i & 7];
}

// ---------------- launcher ----------------
extern "C" void kernel_launch(void* const* d_in, const int* in_sizes, int n_in,
                              void* d_out, int out_size, void* d_ws, size_t ws_size,
                              hipStream_t stream) {
    const float* x    = (const float*)d_in[0];
    const int*   ei   = (const int*)d_in[1];
    const float* ea   = (const float*)d_in[2];
    const float* W1   = (const float*)d_in[3];
    const float* We1  = (const float*)d_in[4];
    const float* as1  = (const float*)d_in[5];
    const float* ad1  = (const float*)d_in[6];
    const float* ae1  = (const float*)d_in[7];
    const float* b1   = (const float*)d_in[8];
    const float* W2   = (const float*)d_in[9];
    const float* We2  = (const float*)d_in[10];
    const float* as2  = (const float*)d_in[11];
    const float* ad2  = (const float*)d_in[12];
    const float* ae2  = (const float*)d_in[13];
    const float* b2   = (const float*)d_in[14];

    const int N  = in_sizes[0] / 64;        // 50000
    const int E  = in_sizes[2];             // 800000
    const int Et = E + N;                   // with self loops
    const int* srcA = ei;
    const int* dstA = ei + E;

    float* ws = (float*)d_ws;
    size_t o = 0;
    float* cnt    = ws + o; o += (size_t)N;
    float* loopa  = ws + o; o += (size_t)N;       // sum -> mean in place
    float* h1     = ws + o; o += (size_t)N * 128;
    float* asrc1  = ws + o; o += (size_t)N * 8;
    float* adst1  = ws + o; o += (size_t)N * 8;
    float* amax1  = ws + o; o += (size_t)N * 8;
    float* den1   = ws + o; o += (size_t)N * 8;
    float* alpha1 = ws + o; o += (size_t)Et * 8;
    float* acc1   = ws + o; o += (size_t)N * 128;
    float* h2     = ws + o; o += (size_t)N * 16;
    float* g2     = ws + o; o += (size_t)N * 8;
    float* asrc2  = ws + o; o += (size_t)N;
    float* adst2  = ws + o; o += (size_t)N;
    float* amax2  = ws + o; o += (size_t)N;
    float* den2   = ws + o; o += (size_t)N;
    float* alpha2 = ws + o; o += (size_t)Et;
    float* acc2   = ws + o; o += (size_t)N * 8;
    float* ce     = ws + o; o += 9;

    const int B = 256;
    auto nb = [](int n, int b) { return (n + b - 1) / b; };

    // self-loop mean edge attr
    fill_f32<<<nb(2 * N, B), B, 0, stream>>>(cnt, 0.f, 2 * N);   // cnt + loopa contiguous
    loop_accum<<<nb(E, B), B, 0, stream>>>(dstA, ea, cnt, loopa, E);
    loop_final<<<nb(N, B), B, 0, stream>>>(loopa, cnt, N);

    // layer 1 dense GEMM via WMMA
    dim3 gg(128 / 16, (N + 15) / 16);
    gemm_wmma_f32<<<gg, 32, 0, stream>>>(x, W1, h1, N);

    node_att1<<<nb(N * 8, B), B, 0, stream>>>(h1, as1, ad1, asrc1, adst1, N);
    edge_consts<<<1, 16, 0, stream>>>(We1, ae1, We2, ae2, ce);

    // layer 1 softmax + aggregate
    fill_f32<<<nb(N * 8, B), B, 0, stream>>>(amax1, -INFINITY, N * 8);
    fill_f32<<<nb(N * 8, B), B, 0, stream>>>(den1, 0.f, N * 8);
    edge_alpha1<<<nb(Et, B), B, 0, stream>>>(srcA, dstA, ea, loopa, asrc1, adst1, ce,
                                             alpha1, amax1, E, Et);
    edge_exp1<<<nb(Et, B), B, 0, stream>>>(dstA, alpha1, amax1, den1, E, Et);
    fill_f32<<<nb(N * 128, B), B, 0, stream>>>(acc1, 0.f, N * 128);
    edge_agg1<<<Et, 128, 0, stream>>>(srcA, dstA, alpha1, den1, h1, acc1, E);
    node_out1<<<nb(N * 16, B), B, 0, stream>>>(acc1, b1, h2, N);

    // layer 2
    node_l2<<<nb(N, B), B, 0, stream>>>(h2, W2, as2, ad2, g2, asrc2, adst2, N);
    fill_f32<<<nb(N, B), B, 0, stream>>>(amax2, -INFINITY, N);
    fill_f32<<<nb(N, B), B, 0, stream>>>(den2, 0.f, N);
    fill_f32<<<nb(N * 8, B), B, 0, stream>>>(acc2, 0.f, N * 8);
    edge_alpha2<<<nb(Et, B), B, 0, stream>>>(srcA, dstA, ea, loopa, asrc2, adst2, ce,
                                             alpha2, amax2, E, Et);
    edge_exp2<<<nb(Et, B), B, 0, stream>>>(dstA, alpha2, amax2, den2, E, Et);
    edge_agg2<<<nb(Et * 8, B), B, 0, stream>>>(srcA, dstA, alpha2, den2, g2, acc2, E, Et);
    final_out<<<nb(N * 8, B), B, 0, stream>>>(acc2, b2, (float*)d_out, N);
}